// PositionalTopDownHTMM_77867757076866
// MI455X (gfx1250) — compile-verified
//
#include <hip/hip_runtime.h>
#include <math.h>

// ---- static topology constants (match reference) ----
#define CDIM   8
#define GDIM   8
#define MDIM   256
#define TDEPTH 10
#define NNODES 1023      // 2^10 - 1
#define NTREES 512
#define LEAF0  511       // first leaf (2^9 - 1)

typedef float v2f __attribute__((ext_vector_type(2)));
typedef float v8f __attribute__((ext_vector_type(8)));

// LDS layout (floats): prior/ratio [1023][64] | A [8][8][2][8] | denomB [8][8] | ll [8]
#define LDS_PRIOR_F (NNODES * 64)
#define LDS_A_F     1024
#define LDS_DENB_F  64
#define LDS_LL_F    8
#define LDS_TOTAL_F (LDS_PRIOR_F + LDS_A_F + LDS_DENB_F + LDS_LL_F)

__global__ __launch_bounds__(32)
void htmm_tree_kernel(const float* __restrict__ lamA,
                      const float* __restrict__ lamB,
                      const float* __restrict__ lamPi,
                      const int*   __restrict__ x,
                      float* __restrict__ out)
{
    extern __shared__ float lds[];
    float* prior = lds;                      // per-node 64 floats: [c*8+g]; later holds ratio
    float* Atab  = lds + LDS_PRIOR_F;        // A[d][c][l][g] = ((d*8+c)*2+l)*8+g
    float* denB  = Atab + LDS_A_F;           // softmax denominators of B, [c*8+g]
    float* llacc = denB + LDS_DENB_F;        // per-tree log-likelihood accumulators [g]

    const int  t        = threadIdx.x;       // 0..31 (wave32)
    const int  tree     = blockIdx.x;
    const long nodeBase = (long)tree * NNODES;
    const int  half     = t >> 4;            // WMMA K-half (lanes 0-15 vs 16-31)
    const int  mn       = t & 15;            // row (A-op) / column (B-op, D)
    const int  n_d      = mn & 7;            // child/parent state packed in N
    const int  n_l      = mn >> 3;           // position packed in N

    // ---------------- parameter normalization ----------------
    // A = softmax over axis 0 (child state d), per (c,l,g): 128 combos, 4/lane
    for (int i = t; i < 128; i += 32) {
        int g = i & 7, l = (i >> 3) & 1, c = i >> 4;
        float e[8], s = 0.f;
#pragma unroll
        for (int d = 0; d < 8; ++d) { e[d] = __expf(lamA[((d*8 + c)*2 + l)*8 + g]); s += e[d]; }
        float inv = 1.f / s;
#pragma unroll
        for (int d = 0; d < 8; ++d) Atab[((d*8 + c)*2 + l)*8 + g] = e[d] * inv;
    }
    // B softmax denominators over axis 1 (m), per (c,g)
    for (int i = t; i < 64; i += 32) {
        int g = i & 7, c = i >> 3;
        float s = 0.f;
        for (int m = 0; m < MDIM; ++m) s += __expf(lamB[(c*MDIM + m)*8 + g]);
        denB[c*8 + g] = s;
    }
    // Pi softmax over axis 0 -> root prior; init ll accumulators
    if (t < 8) {
        int g = t;
        float e[8], s = 0.f;
#pragma unroll
        for (int c = 0; c < 8; ++c) { e[c] = __expf(lamPi[c*8 + g]); s += e[c]; }
        float inv = 1.f / s;
#pragma unroll
        for (int c = 0; c < 8; ++c) prior[c*8 + g] = e[c] * inv;
        llacc[g] = 0.f;
    }
    __syncthreads();

    // ---------------- downward pass: priors ----------------
    // pr[p,l,d,g] = sum_c prior[pa=p][c,g] * A[d,c,l,g]
    // WMMA: M=16 parents, N=16=(l<<3|d), K=8 parent states as 2x K=4 f32 WMMAs.
    for (int d = 0; d < TDEPTH - 1; ++d) {
        const int P  = 1 << d;
        const int P0 = P - 1;
        const int CB = 2 * P - 1;            // first child node of this level
        const int ntile = (P + 15) >> 4;
        for (int tile = 0; tile < ntile; ++tile) {
            const int pt = tile << 4;
            int paA = pt + mn; if (paA > P - 1) paA = P - 1;   // clamp (pad rows read valid data)
            const int pnode = P0 + paA;
            for (int g = 0; g < 8; ++g) {
                v8f acc = {0.f, 0.f, 0.f, 0.f, 0.f, 0.f, 0.f, 0.f};
#pragma unroll
                for (int kb = 0; kb < 8; kb += 4) {
                    const int c0 = kb + half * 2;              // K layout: lanes0-15->K0,1 lanes16-31->K2,3
                    v2f a, b;
                    a.x = prior[pnode*64 + (c0    )*8 + g];
                    a.y = prior[pnode*64 + (c0 + 1)*8 + g];
                    b.x = Atab[((n_d*8 + c0    )*2 + n_l)*8 + g];
                    b.y = Atab[((n_d*8 + c0 + 1)*2 + n_l)*8 + g];
                    acc = __builtin_amdgcn_wmma_f32_16x16x4_f32(false, a, false, b,
                                                                (short)0, acc, false, false);
                }
                // D: vgpr r holds row M = half*8 + r, col (n_l, n_d)
#pragma unroll
                for (int r = 0; r < 8; ++r) {
                    const int prow = pt + half*8 + r;
                    if (prow < P)
                        prior[(CB + 2*prow + n_l)*64 + n_d*8 + g] = acc[r];
                }
            }
        }
        __syncthreads();
    }

    // ---------------- leaf emissions ----------------
    // beta = prior*Bx ; nu = sum_c ; ratio(leaf) = Bx/nu stored over prior ; ll += log(nu)
    {
        const int g  = t & 7;
        const int cg = t >> 3;               // this lane handles c in {cg, cg+4}
        const float db0 = denB[(cg    )*8 + g];
        const float db1 = denB[(cg + 4)*8 + g];
        for (int leaf = LEAF0; leaf < NNODES; ++leaf) {
            const int   xn  = x[nodeBase + leaf];
            const float bx0 = __expf(lamB[((cg    )*MDIM + xn)*8 + g]) / db0;
            const float bx1 = __expf(lamB[((cg + 4)*MDIM + xn)*8 + g]) / db1;
            const float pb0 = prior[leaf*64 + (cg    )*8 + g] * bx0;
            const float pb1 = prior[leaf*64 + (cg + 4)*8 + g] * bx1;
            float nu = pb0 + pb1;
            nu += __shfl_xor(nu, 8, 32);     // reduce across the 4 c-groups (lane bits 3,4)
            nu += __shfl_xor(nu, 16, 32);
            const float inv = 1.f / nu;
            prior[leaf*64 + (cg    )*8 + g] = bx0 * inv;   // ratio = beta/prior = Bx/nu
            prior[leaf*64 + (cg + 4)*8 + g] = bx1 * inv;
            if (t < 8) llacc[g] += __logf(nu);             // unique g per lane 0-7
        }
    }
    __syncthreads();

    // ---------------- upward pass ----------------
    // beta_uv[p,l',d_pa,g] = sum_c ratio[p,l',c,g] * A[c,d_pa,l',g]
    // K=16 over k=(l<<3|c); B zero-masked where l(k)!=l'(col): 4x K=4 f32 WMMAs.
    for (int d = TDEPTH - 2; d >= 0; --d) {
        const int P  = 1 << d;
        const int P0 = P - 1;
        const int CB = 2 * P - 1;
        const int ntile = (P + 15) >> 4;
        for (int tile = 0; tile < ntile; ++tile) {
            const int pt = tile << 4;
            int paA = pt + mn; if (paA > P - 1) paA = P - 1;
            int pnv[8], xv[8];
#pragma unroll
            for (int r = 0; r < 8; ++r) {
                int prow = pt + half*8 + r; if (prow > P - 1) prow = P - 1;
                pnv[r] = P0 + prow;
                xv[r]  = x[nodeBase + pnv[r]];
            }
            for (int g = 0; g < 8; ++g) {
                v8f acc = {0.f, 0.f, 0.f, 0.f, 0.f, 0.f, 0.f, 0.f};
#pragma unroll
                for (int kb = 0; kb < 16; kb += 4) {
                    const int k0  = kb + half * 2;
                    const int l0  = (k0 >> 3) & 1, c0v = k0 & 7;
                    const int k1  = k0 + 1;
                    const int l1  = (k1 >> 3) & 1, c1v = k1 & 7;
                    v2f a, b;
                    a.x = prior[(CB + 2*paA + l0)*64 + c0v*8 + g];   // child ratios
                    a.y = prior[(CB + 2*paA + l1)*64 + c1v*8 + g];
                    b.x = (l0 == n_l) ? Atab[((c0v*8 + n_d)*2 + l0)*8 + g] : 0.f;
                    b.y = (l1 == n_l) ? Atab[((c1v*8 + n_d)*2 + l1)*8 + g] : 0.f;
                    acc = __builtin_amdgcn_wmma_f32_16x16x4_f32(false, a, false, b,
                                                                (short)0, acc, false, false);
                }
#pragma unroll
                for (int r = 0; r < 8; ++r) {
                    float v = acc[r];
                    const float prodv = v * __shfl_xor(v, 8, 32);     // product over l (lane bit 3)
                    const int   prow  = pt + half*8 + r;
                    const bool  valid = (prow < P);
                    const int   pn    = pnv[r];
                    const float pr    = prior[pn*64 + n_d*8 + g];
                    const float bx    = __expf(lamB[(n_d*MDIM + xv[r])*8 + g]) / denB[n_d*8 + g];
                    const float pb    = pr * bx;
                    float nu = pb * pb * prodv;                       // unnorm = (prior*Bx)^2 * prod_ch
                    nu += __shfl_xor(nu, 1, 32);                      // sum over d_pa (8-lane groups)
                    nu += __shfl_xor(nu, 2, 32);
                    nu += __shfl_xor(nu, 4, 32);
                    const float ratio = pr * bx * bx * prodv / nu;    // beta_new/prior
                    if (valid && n_l == 0) {
                        prior[pn*64 + n_d*8 + g] = ratio;             // overwrite prior with ratio
                        if (n_d == 0) atomicAdd(&llacc[g], __logf(nu)); // lanes 0 & 16 -> ds_add_f32
                    }
                }
            }
        }
        __syncthreads();
    }

    // ---------------- per-tree output ----------------
    if (t < 8) out[tree*8 + t] = llacc[t];
}

extern "C" void kernel_launch(void* const* d_in, const int* in_sizes, int n_in,
                              void* d_out, int out_size, void* d_ws, size_t ws_size,
                              hipStream_t stream)
{
    (void)in_sizes; (void)n_in; (void)out_size; (void)d_ws; (void)ws_size;
    const float* lamA  = (const float*)d_in[0];   // (8,8,2,8)
    const float* lamB  = (const float*)d_in[1];   // (8,256,8)
    const float* lamPi = (const float*)d_in[2];   // (8,8)
    const int*   x     = (const int*)  d_in[3];   // (523776,)
    float* out = (float*)d_out;                   // (512,8)

    const size_t ldsBytes = (size_t)LDS_TOTAL_F * sizeof(float);  // ~266 KB < 320 KB/WGP
    (void)hipFuncSetAttribute((const void*)htmm_tree_kernel,
                              hipFuncAttributeMaxDynamicSharedMemorySize, (int)ldsBytes);
    htmm_tree_kernel<<<NTREES, 32, ldsBytes, stream>>>(lamA, lamB, lamPi, x, out);
}